// AttentionBlock_79963701117176
// MI455X (gfx1250) — compile-verified
//
#include <hip/hip_runtime.h>
#include <math.h>

typedef __attribute__((ext_vector_type(2))) float v2f;
typedef __attribute__((ext_vector_type(8))) float v8f;

#define C_DIM 2048
#define K_DIM 256
#define B_DIM 4096

// ---------------------------------------------------------------------------
// Stage 1: theta = x@Wt + bt, phi = x@Wp + bp, f = x@Wf + bf   ([B,C]x[C,K])
// One wave computes one 16x16 tile of ALL THREE outputs (A fragment reused 3x).
// V_WMMA_F32_16X16X4_F32: A = 16x4 (2 VGPRs), B = 4x16 (2 VGPRs), C/D = 8 VGPRs.
// ---------------------------------------------------------------------------
__global__ __launch_bounds__(256) void qkf_gemm_kernel(
    const float* __restrict__ x,
    const float* __restrict__ Wt, const float* __restrict__ bt,
    const float* __restrict__ Wp, const float* __restrict__ bp,
    const float* __restrict__ Wf, const float* __restrict__ bf,
    float* __restrict__ theta, float* __restrict__ phi, float* __restrict__ f)
{
    const int lane  = threadIdx.x & 31;
    const int wave  = threadIdx.x >> 5;
    const int tile  = blockIdx.x * 8 + wave;   // (B/16)*(K/16) = 256*16 = 4096 tiles
    const int row0  = (tile >> 4) * 16;        // over B
    const int col0  = (tile & 15) * 16;        // over K
    const int lrow  = lane & 15;
    const int khalf = (lane >> 4) << 1;        // lanes 0-15: K+0/1, lanes 16-31: K+2/3
    const int coln  = col0 + lrow;

    // Accumulators initialized with bias (bias is per-column => same in all 8 VGPRs)
    const float btv = bt[coln], bpv = bp[coln], bfv = bf[coln];
    v8f ct, cp, cf;
#pragma unroll
    for (int i = 0; i < 8; ++i) { ct[i] = btv; cp[i] = bpv; cf[i] = bfv; }

    const float* xrow = x + (size_t)(row0 + lrow) * C_DIM;

    for (int k = 0; k < C_DIM; k += 4) {
        // A: 16x4 tile of x — one b64 load per lane
        v2f a = *(const v2f*)(xrow + k + khalf);
        // B: 4x16 tiles of the three weight matrices (row k in low lanes half,
        // row k+2 in high lanes half; VGPR0/1 = consecutive K rows)
        const float* wtp = Wt + (size_t)(k + khalf) * K_DIM + coln;
        const float* wpp = Wp + (size_t)(k + khalf) * K_DIM + coln;
        const float* wfp = Wf + (size_t)(k + khalf) * K_DIM + coln;
        v2f b0, b1, b2;
        b0.x = wtp[0]; b0.y = wtp[K_DIM];
        b1.x = wpp[0]; b1.y = wpp[K_DIM];
        b2.x = wfp[0]; b2.y = wfp[K_DIM];
        ct = __builtin_amdgcn_wmma_f32_16x16x4_f32(false, a, false, b0, (short)0, ct, false, false);
        cp = __builtin_amdgcn_wmma_f32_16x16x4_f32(false, a, false, b1, (short)0, cp, false, false);
        cf = __builtin_amdgcn_wmma_f32_16x16x4_f32(false, a, false, b2, (short)0, cf, false, false);
    }

    // D layout: VGPR v, lanes 0-15 -> row row0+v, lanes 16-31 -> row row0+v+8
    const int rbase = row0 + ((lane >> 4) << 3);
#pragma unroll
    for (int v = 0; v < 8; ++v) {
        size_t off = (size_t)(rbase + v) * K_DIM + coln;
        theta[off] = ct[v];
        phi[off]   = cp[v];
        f[off]     = cf[v];
    }
}

// ---------------------------------------------------------------------------
// Stage 2: t[b,i] = sum_j softmax_j(theta_i*phi_j) * f_j  (rank-1 scores, never
// materialized). One block per sample; phi/f live in LDS (broadcast reads).
// max_j(theta_i*phi_j) = theta_i>=0 ? theta_i*max(phi) : theta_i*min(phi).
// This kernel carries the 268M v_exp_f32 TRANS workload.
// ---------------------------------------------------------------------------
__global__ __launch_bounds__(256) void softmax_av_kernel(
    const float* __restrict__ theta, const float* __restrict__ phi,
    const float* __restrict__ f, float* __restrict__ t)
{
    __shared__ float sphi[K_DIM];
    __shared__ float sf[K_DIM];
    __shared__ float rmax[K_DIM];
    __shared__ float rmin[K_DIM];

    const int b = blockIdx.x;
    const int i = threadIdx.x;

    const float ph = phi[(size_t)b * K_DIM + i];
    sphi[i] = ph;
    sf[i]   = f[(size_t)b * K_DIM + i];
    rmax[i] = ph;
    rmin[i] = ph;
    __syncthreads();

    // Block tree-reduction for max/min of phi (exact softmax max subtraction)
#pragma unroll
    for (int s = 128; s > 0; s >>= 1) {
        if (i < s) {
            rmax[i] = fmaxf(rmax[i], rmax[i + s]);
            rmin[i] = fminf(rmin[i], rmin[i + s]);
        }
        __syncthreads();
    }
    const float phimax = rmax[0];
    const float phimin = rmin[0];

    const float th = theta[(size_t)b * K_DIM + i];
    const float m  = (th >= 0.0f) ? th * phimax : th * phimin;

    float den = 0.0f, num = 0.0f;
#pragma unroll 8
    for (int j = 0; j < K_DIM; ++j) {
        const float s = __expf(fmaf(th, sphi[j], -m));
        den += s;
        num = fmaf(s, sf[j], num);
    }
    t[(size_t)b * K_DIM + i] = num / den;
}

// ---------------------------------------------------------------------------
// Stage 3: out = x + t@Wg + bg   ([B,K]x[K,C] + residual)
// ---------------------------------------------------------------------------
__global__ __launch_bounds__(256) void out_gemm_kernel(
    const float* __restrict__ t, const float* __restrict__ Wg,
    const float* __restrict__ bg, const float* __restrict__ x,
    float* __restrict__ out)
{
    const int lane  = threadIdx.x & 31;
    const int wave  = threadIdx.x >> 5;
    const int tile  = blockIdx.x * 8 + wave;   // (B/16)*(C/16) = 256*128 = 32768
    const int row0  = (tile >> 7) * 16;        // over B
    const int col0  = (tile & 127) * 16;       // over C
    const int lrow  = lane & 15;
    const int khalf = (lane >> 4) << 1;
    const int coln  = col0 + lrow;

    const float bgv = bg[coln];
    v8f acc;
#pragma unroll
    for (int i = 0; i < 8; ++i) acc[i] = bgv;

    const float* trow = t + (size_t)(row0 + lrow) * K_DIM;

#pragma unroll 4
    for (int k = 0; k < K_DIM; k += 4) {
        v2f a = *(const v2f*)(trow + k + khalf);
        const float* wp = Wg + (size_t)(k + khalf) * C_DIM + coln;
        v2f bm;
        bm.x = wp[0]; bm.y = wp[C_DIM];
        acc = __builtin_amdgcn_wmma_f32_16x16x4_f32(false, a, false, bm, (short)0, acc, false, false);
    }

    const int rbase = row0 + ((lane >> 4) << 3);
#pragma unroll
    for (int v = 0; v < 8; ++v) {
        size_t off = (size_t)(rbase + v) * C_DIM + coln;
        out[off] = acc[v] + x[off];
    }
}

// ---------------------------------------------------------------------------
extern "C" void kernel_launch(void* const* d_in, const int* in_sizes, int n_in,
                              void* d_out, int out_size, void* d_ws, size_t ws_size,
                              hipStream_t stream) {
    const float* x  = (const float*)d_in[0];
    const float* Wt = (const float*)d_in[1];
    const float* bt = (const float*)d_in[2];
    const float* Wp = (const float*)d_in[3];
    const float* bp = (const float*)d_in[4];
    const float* Wf = (const float*)d_in[5];
    const float* bf = (const float*)d_in[6];
    const float* Wg = (const float*)d_in[7];
    const float* bg = (const float*)d_in[8];
    float* out = (float*)d_out;

    // Workspace layout: theta | phi | f | t   (each B*K f32 = 4 MB, 16 MB total)
    float* theta = (float*)d_ws;
    float* phi   = theta + (size_t)B_DIM * K_DIM;
    float* f     = phi   + (size_t)B_DIM * K_DIM;
    float* t     = f     + (size_t)B_DIM * K_DIM;

    // Stage 1: 4096 tiles / 8 waves per block = 512 blocks
    qkf_gemm_kernel<<<512, 256, 0, stream>>>(x, Wt, bt, Wp, bp, Wf, bf, theta, phi, f);
    // Stage 2: one block per sample
    softmax_av_kernel<<<B_DIM, 256, 0, stream>>>(theta, phi, f, t);
    // Stage 3: 32768 tiles / 8 waves per block = 4096 blocks
    out_gemm_kernel<<<4096, 256, 0, stream>>>(t, Wg, bg, x, out);
}